// LaneAttention_LSTM_25675314495804
// MI455X (gfx1250) — compile-verified
//
#include <hip/hip_runtime.h>

typedef float v2f __attribute__((ext_vector_type(2)));
typedef float v8f __attribute__((ext_vector_type(8)));

#define M_TOTAL   262144
#define NGROUPS   32768
#define ROWS_PER_WAVE 16
#define WAVES_PER_BLOCK 8

// Fused: per-lane MLP score (f32 WMMA) -> per-group softmax -> weighted pooling.
// Group = 8 consecutive rows; one wave owns 16 rows = 2 whole groups.
__global__ __launch_bounds__(256)
void lane_attention_pool_kernel(
    const float* __restrict__ lane_ht,      // [M,64]
    const float* __restrict__ lane_info,    // [M,64]
    const float* __restrict__ lane_future,  // [M,64]
    const float* __restrict__ W1,           // [128,16]
    const float* __restrict__ b1,           // [16]
    const float* __restrict__ W2,           // [16,1]
    const float* __restrict__ b2,           // [1]
    const int*   __restrict__ seg,          // [M,1] sorted group ids
    float*       __restrict__ out)          // [N,192]
{
    __shared__ float sW1[128 * 16];
    const int tid = threadIdx.x;
    for (int i = tid; i < 128 * 16; i += 256) sW1[i] = W1[i];
    __syncthreads();

    const int lane = tid & 31;
    const int wave = tid >> 5;
    const int tile = blockIdx.x * WAVES_PER_BLOCK + wave; // 16 rows per tile
    const int row0 = tile * ROWS_PER_WAVE;
    const int n    = lane & 15;  // A: row index / C: column index
    const int hi   = lane >> 4;  // half-wave select

    // ---- h_pre = x @ W1 via V_WMMA_F32_16X16X4_F32, K=128 in 32 steps ----
    // A layout (16x4 f32): lane holds row n, K = 2*hi + {0,1}
    // B layout (4x16 f32): lane holds col n, K = 2*hi + {0,1}
    v8f c = {};
    #pragma unroll 4
    for (int k = 0; k < 32; ++k) {
        const float* srcA = (k < 16) ? lane_ht : lane_info;
        const int colA = 4 * (k & 15) + 2 * hi;
        const float2 av = *reinterpret_cast<const float2*>(
            srcA + (size_t)(row0 + n) * 64 + colA);
        v2f a; a[0] = av.x; a[1] = av.y;
        const int kr = 4 * k + 2 * hi;
        v2f b; b[0] = sW1[kr * 16 + n]; b[1] = sW1[(kr + 1) * 16 + n];
        c = __builtin_amdgcn_wmma_f32_16x16x4_f32(
            /*neg_a=*/false, a, /*neg_b=*/false, b,
            /*c_mod=*/(short)0, c, /*reuse_a=*/false, /*reuse_b=*/false);
    }

    // ---- bias + ReLU + (h @ W2 + b2); reduce 16 lanes -> score per row ----
    // C layout: lane holds h_pre[v + 8*hi][n] in c[v]
    const float b1n = b1[n];
    const float w2n = W2[n];
    const float b2s = b2[0];
    float s[8];
    #pragma unroll
    for (int v = 0; v < 8; ++v) {
        float hp = c[v] + b1n;
        float t  = (hp > 0.f ? hp : 0.f) * w2n;
        t += __shfl_xor(t, 1, 32);
        t += __shfl_xor(t, 2, 32);
        t += __shfl_xor(t, 4, 32);
        t += __shfl_xor(t, 8, 32);
        s[v] = t + b2s;              // score of row (row0 + v + 8*hi)
    }

    // ---- per-group softmax: each lane holds all 8 scores of its group ----
    float mx = s[0];
    #pragma unroll
    for (int v = 1; v < 8; ++v) mx = fmaxf(mx, s[v]);
    float p[8], denom = 0.f;
    #pragma unroll
    for (int v = 0; v < 8; ++v) { p[v] = __expf(s[v] - mx); denom += p[v]; }
    const float inv = 1.f / denom;
    // share probs across half-waves: pA = group (rows 0-7), pB = group (rows 8-15)
    float pA[8], pB[8];
    #pragma unroll
    for (int v = 0; v < 8; ++v) {
        const float own = p[v] * inv;
        const float oth = __shfl_xor(own, 16, 32);
        pA[v] = hi ? oth : own;
        pB[v] = hi ? own : oth;
    }

    // ---- softmax-weighted pooling of [ht|info|future] into the 2 group rows ----
    const int gA = seg[row0];
    const int gB = seg[row0 + 8];
    const float* srcs[3] = { lane_ht, lane_info, lane_future };
    #pragma unroll
    for (int sIdx = 0; sIdx < 3; ++sIdx) {
        const float* src = srcs[sIdx];
        float aA0 = 0.f, aA1 = 0.f, aB0 = 0.f, aB1 = 0.f;
        #pragma unroll
        for (int r = 0; r < 8; ++r) {
            const float* rp = src + (size_t)(row0 + r) * 64;
            aA0 += pA[r] * rp[lane];
            aA1 += pA[r] * rp[lane + 32];
            const float* rq = src + (size_t)(row0 + 8 + r) * 64;
            aB0 += pB[r] * rq[lane];
            aB1 += pB[r] * rq[lane + 32];
        }
        float* oA = out + (size_t)gA * 192 + sIdx * 64;
        float* oB = out + (size_t)gB * 192 + sIdx * 64;
        oA[lane]      = aA0;
        oA[lane + 32] = aA1;
        oB[lane]      = aB0;
        oB[lane + 32] = aB1;
    }
}

extern "C" void kernel_launch(void* const* d_in, const int* in_sizes, int n_in,
                              void* d_out, int out_size, void* d_ws, size_t ws_size,
                              hipStream_t stream) {
    const float* lane_ht     = (const float*)d_in[0];
    const float* lane_info   = (const float*)d_in[1];
    const float* lane_future = (const float*)d_in[2];
    const float* W1          = (const float*)d_in[3];
    const float* b1          = (const float*)d_in[4];
    const float* W2          = (const float*)d_in[5];
    const float* b2          = (const float*)d_in[6];
    const int*   seg         = (const int*)d_in[7];
    float*       out         = (float*)d_out;

    const int tiles  = M_TOTAL / ROWS_PER_WAVE;          // 16384 waves
    const int blocks = tiles / WAVES_PER_BLOCK;          // 2048 blocks
    lane_attention_pool_kernel<<<dim3(blocks), dim3(256), 0, stream>>>(
        lane_ht, lane_info, lane_future, W1, b1, W2, b2, seg, out);
}